// GATSkip_WeightShare_27092653703873
// MI455X (gfx1250) — compile-verified
//
#include <hip/hip_runtime.h>
#include <math.h>

#define N_NODES 100000
#define DIM 64
#define N_EDGES 1200000
#define E_TOTAL (N_EDGES + N_NODES)
#define SLOPE 0.2f
#define LDS_PITCH 72   // 2*72 = 144 ≡ 16 (mod 64): lane-halves hit disjoint bank sets

typedef __attribute__((ext_vector_type(2))) float v2f;
typedef __attribute__((ext_vector_type(8))) float v8f;

// ---------------- order-preserving float<->uint encoding for atomicMax ------
__device__ __forceinline__ unsigned fenc(float f) {
  unsigned u = __float_as_uint(f);
  return (u & 0x80000000u) ? ~u : (u | 0x80000000u);
}
__device__ __forceinline__ float fdec(unsigned u) {
  return __uint_as_float((u & 0x80000000u) ? (u ^ 0x80000000u) : ~u);
}
__device__ __forceinline__ float lrelu(float v) { return v > 0.f ? v : SLOPE * v; }

__device__ __forceinline__ void edge_sd(const int* __restrict__ src,
                                        const int* __restrict__ dst,
                                        int e, int& s, int& d) {
  if (e < N_EDGES) { s = src[e]; d = dst[e]; }
  else             { s = e - N_EDGES; d = s; }   // self loops
}

// ---------------- H[n,64] = X[n,64] @ W[64,64]  (+= if accumulate) ----------
// W staged in LDS once per block (bank-conflict-free padded pitch); one wave
// per 16-row tile; V_WMMA_F32_16X16X4_F32 with A-fragment reuse across the 4
// column tiles.
__global__ __launch_bounds__(256) void gemm64_wmma(const float* __restrict__ X,
                                                   const float* __restrict__ W,
                                                   float* __restrict__ H,
                                                   int nTiles, int accumulate) {
  __shared__ float Wl[DIM * LDS_PITCH];

  // Stage W (64x64 f32 = 16KB): 1024 float4 across 256 threads.
  {
    const float4* W4 = (const float4*)W;
#pragma unroll
    for (int i = 0; i < 4; ++i) {
      const int idx4 = threadIdx.x + i * 256;
      const int flat = idx4 * 4;
      const int row  = flat >> 6;
      const int col  = flat & 63;
      const float4 w = W4[idx4];
      float* p = &Wl[row * LDS_PITCH + col];
      p[0] = w.x; p[1] = w.y; p[2] = w.z; p[3] = w.w;
    }
  }
  __syncthreads();

  const int lane = threadIdx.x & 31;
  const int wave = threadIdx.x >> 5;
  const int tile = blockIdx.x * 8 + wave;
  if (tile >= nTiles) return;              // wave-uniform: EXEC stays all-ones
  const int row0 = tile << 4;
  const int m  = lane & 15;                // A row / B,C,D column
  const int hi = lane >> 4;                // A: K-pair select; C/D: row-half

  // Preload 16 A fragments (K=64 -> 16 steps of K=4) as float2 (b64 loads).
  v2f a[16];
  const float2* xrow2 = (const float2*)(X + (size_t)(row0 + m) * DIM);
#pragma unroll
  for (int kk = 0; kk < 16; ++kk) {
    const float2 av = xrow2[kk * 2 + hi];  // floats [kk*4+2*hi, +1]
    a[kk].x = av.x;
    a[kk].y = av.y;
  }

#pragma unroll
  for (int ct = 0; ct < 4; ++ct) {
    const int col0 = ct * 16;
    v8f c;
    if (accumulate) {
#pragma unroll
      for (int v = 0; v < 8; ++v)
        c[v] = H[(size_t)(row0 + v + 8 * hi) * DIM + col0 + m];
    } else {
#pragma unroll
      for (int v = 0; v < 8; ++v) c[v] = 0.0f;
    }
#pragma unroll
    for (int kk = 0; kk < 16; ++kk) {
      const int kb = kk * 4 + 2 * hi;
      v2f b;
      b.x = Wl[(kb + 0) * LDS_PITCH + col0 + m];
      b.y = Wl[(kb + 1) * LDS_PITCH + col0 + m];
      c = __builtin_amdgcn_wmma_f32_16x16x4_f32(false, a[kk], false, b,
                                                (short)0, c, false, false);
    }
#pragma unroll
    for (int v = 0; v < 8; ++v)
      H[(size_t)(row0 + v + 8 * hi) * DIM + col0 + m] = c[v];
  }
}

// ---------------- per-node logit parts + softmax state reset ----------------
__global__ __launch_bounds__(256) void node_prep(const float* __restrict__ H,
                                                 const float* __restrict__ a_src,
                                                 const float* __restrict__ a_dst,
                                                 float* __restrict__ al,
                                                 float* __restrict__ ar,
                                                 unsigned* __restrict__ m_enc,
                                                 float* __restrict__ z,
                                                 float* __restrict__ acc) {
  const int lane = threadIdx.x & 31;
  const int wave = threadIdx.x >> 5;
  const int row  = blockIdx.x * 8 + wave;
  if (row >= N_NODES) return;
  const float2 h  = ((const float2*)(H + (size_t)row * DIM))[lane];
  const float2 as = ((const float2*)a_src)[lane];
  const float2 ad = ((const float2*)a_dst)[lane];
  float pl = h.x * as.x + h.y * as.y;
  float pr = h.x * ad.x + h.y * ad.y;
#pragma unroll
  for (int off = 16; off > 0; off >>= 1) {
    pl += __shfl_xor(pl, off, 32);
    pr += __shfl_xor(pr, off, 32);
  }
  float2 zz; zz.x = 0.f; zz.y = 0.f;
  ((float2*)(acc + (size_t)row * DIM))[lane] = zz;   // zero accumulator row
  if (lane == 0) {
    al[row]    = pl;
    ar[row]    = pr;
    m_enc[row] = 0x007FFFFFu;  // fenc(-inf)
    z[row]     = 0.f;
  }
}

// ---------------- softmax pass 1: per-destination max ----------------------
__global__ __launch_bounds__(256) void edge_max(const int* __restrict__ src,
                                                const int* __restrict__ dst,
                                                const float* __restrict__ al,
                                                const float* __restrict__ ar,
                                                unsigned* __restrict__ m_enc) {
  const int e = blockIdx.x * blockDim.x + threadIdx.x;
  if (e >= E_TOTAL) return;
  int s, d; edge_sd(src, dst, e, s, d);
  const float logit = lrelu(al[s] + ar[d]);
  atomicMax(&m_enc[d], fenc(logit));
}

// ---------------- softmax pass 2: per-destination sum of exp ---------------
__global__ __launch_bounds__(256) void edge_sum(const int* __restrict__ src,
                                                const int* __restrict__ dst,
                                                const float* __restrict__ al,
                                                const float* __restrict__ ar,
                                                const unsigned* __restrict__ m_enc,
                                                float* __restrict__ z) {
  const int e = blockIdx.x * blockDim.x + threadIdx.x;
  if (e >= E_TOTAL) return;
  int s, d; edge_sd(src, dst, e, s, d);
  const float logit = lrelu(al[s] + ar[d]);
  atomicAdd(&z[d], expf(logit - fdec(m_enc[d])));
}

// ---------------- weighted aggregation: acc[d] += alpha * h[s] -------------
// 64 threads per edge -> coalesced 256B gather/scatter of feature rows.
__global__ __launch_bounds__(256) void edge_agg(const float* __restrict__ H,
                                                const int* __restrict__ src,
                                                const int* __restrict__ dst,
                                                const float* __restrict__ al,
                                                const float* __restrict__ ar,
                                                const unsigned* __restrict__ m_enc,
                                                const float* __restrict__ z,
                                                float* __restrict__ acc) {
  const int idx = blockIdx.x * blockDim.x + threadIdx.x;
  const int e = idx >> 6;
  const int f = idx & 63;
  if (e >= E_TOTAL) return;
  int s, d; edge_sd(src, dst, e, s, d);
  const float logit = lrelu(al[s] + ar[d]);
  const float alpha = expf(logit - fdec(m_enc[d])) / z[d];
  atomicAdd(&acc[(size_t)d * DIM + f], alpha * H[(size_t)s * DIM + f]);
}

// ---------------- bias + (optional) exact-erf GELU -------------------------
__global__ __launch_bounds__(256) void node_finish(const float* __restrict__ acc,
                                                   const float* __restrict__ b,
                                                   float* __restrict__ out,
                                                   int do_gelu) {
  const int i = blockIdx.x * blockDim.x + threadIdx.x;
  if (i >= N_NODES * DIM) return;
  float v = acc[i] + b[i & 63];
  if (do_gelu) v = 0.5f * v * (1.0f + erff(v * 0.70710678118654752f));
  out[i] = v;
}

extern "C" void kernel_launch(void* const* d_in, const int* in_sizes, int n_in,
                              void* d_out, int out_size, void* d_ws, size_t ws_size,
                              hipStream_t stream) {
  (void)in_sizes; (void)n_in; (void)out_size; (void)ws_size;

  const float* x        = (const float*)d_in[0];
  const int*   eidx     = (const int*)d_in[1];
  const float* W_std    = (const float*)d_in[2];
  const float* a_src_s  = (const float*)d_in[3];
  const float* a_dst_s  = (const float*)d_in[4];
  const float* b_std    = (const float*)d_in[5];
  const float* W_skip   = (const float*)d_in[6];
  const float* a_src_k  = (const float*)d_in[7];
  const float* a_dst_k  = (const float*)d_in[8];
  const float* b_skip   = (const float*)d_in[9];
  float* out = (float*)d_out;

  const int* src = eidx;
  const int* dst = eidx + N_EDGES;

  // workspace layout (floats): H[N*64] | bufA[N*64] | al[N] | ar[N] | m_enc[N] | z[N]
  float*    H     = (float*)d_ws;
  float*    bufA  = H + (size_t)N_NODES * DIM;
  float*    al    = bufA + (size_t)N_NODES * DIM;
  float*    ar    = al + N_NODES;
  unsigned* m_enc = (unsigned*)(ar + N_NODES);
  float*    z     = (float*)(m_enc + N_NODES);

  const int nTiles = N_NODES / 16;                        // 6250
  const dim3 gGemm((nTiles + 7) / 8), gNode((N_NODES + 7) / 8);
  const dim3 gEdge((E_TOTAL + 255) / 256);
  const dim3 gAgg(((size_t)E_TOTAL * 64 + 255) / 256);
  const dim3 gFin((N_NODES * DIM + 255) / 256);
  const dim3 blk(256);

  // ---- 4 weight-shared GAT layers (in-place ping in bufA) ----
  for (int l = 0; l < 4; ++l) {
    const float* xin = (l == 0) ? x : bufA;
    gemm64_wmma<<<gGemm, blk, 0, stream>>>(xin, W_std, H, nTiles, 0);
    node_prep  <<<gNode, blk, 0, stream>>>(H, a_src_s, a_dst_s, al, ar, m_enc, z, bufA);
    edge_max   <<<gEdge, blk, 0, stream>>>(src, dst, al, ar, m_enc);
    edge_sum   <<<gEdge, blk, 0, stream>>>(src, dst, al, ar, m_enc, z);
    edge_agg   <<<gAgg,  blk, 0, stream>>>(H, src, dst, al, ar, m_enc, z, bufA);
    node_finish<<<gFin,  blk, 0, stream>>>(bufA, b_std, bufA, /*gelu=*/1);
  }

  // ---- skip layer: h = [x0 | x4] @ W_skip done as two accumulating GEMMs ----
  gemm64_wmma<<<gGemm, blk, 0, stream>>>(x,    W_skip,             H, nTiles, 0);
  gemm64_wmma<<<gGemm, blk, 0, stream>>>(bufA, W_skip + DIM * DIM, H, nTiles, 1);
  node_prep  <<<gNode, blk, 0, stream>>>(H, a_src_k, a_dst_k, al, ar, m_enc, z, bufA);
  edge_max   <<<gEdge, blk, 0, stream>>>(src, dst, al, ar, m_enc);
  edge_sum   <<<gEdge, blk, 0, stream>>>(src, dst, al, ar, m_enc, z);
  edge_agg   <<<gAgg,  blk, 0, stream>>>(H, src, dst, al, ar, m_enc, z, bufA);
  node_finish<<<gFin,  blk, 0, stream>>>(bufA, b_skip, out, /*gelu=*/0);
}